// GramSchmidt02_19980187861910
// MI455X (gfx1250) — compile-verified
//
#include <hip/hip_runtime.h>
#include <hip/hip_bf16.h>
#include <stddef.h>

// Panel-blocked classical Gram-Schmidt for x: [K=2048, L=8192] fp32.
// Bulk work = two fp32 WMMA GEMM families per 16-row panel. The 64MB matrix is
// L2-resident (192MB L2), so the GEMMs bind on the matrix pipe / L2 bandwidth:
//  - full-precision V_WMMA_F32_16X16X4_F32 (no downcast),
//  - register blocking in the update GEMM (1 A-fragment reused for 2 tiles),
//  - L-split partials in the correlation GEMM for 4x wave-level parallelism
//    (deterministic: private partial buffers, no float atomics).

typedef __attribute__((ext_vector_type(2))) float v2f;
typedef __attribute__((ext_vector_type(8))) float v8f;

#define GS_K   2048
#define GS_L   8192
#define PANEL  16
#define LSPLIT 4
#define KSLICE (GS_L / LSPLIT)   // 2048

// ---------------------------------------------------------------------------
// Kernel A: Zpart[s][m][j] = dot(Q[m], Xorig[i0+j]) over L-slice s.
// grid = (i0/16, LSPLIT), one wave per (16-row tile, L-slice).
//   A (16x4):  lane<16: M=lane, v0/v1 = K0/K1 ; lane>=16: M=lane-16, K2/K3
//   B (4x16):  lane<16: N=lane, v0/v1 = K0/K1 ; lane>=16: N=lane-16, K2/K3
//   C/D:       VGPR r -> rows r (lanes 0-15) and 8+r (lanes 16-31)
// ---------------------------------------------------------------------------
__global__ void __launch_bounds__(32)
gs_panel_corr(const float* __restrict__ Q, const float* __restrict__ Xorig,
              float* __restrict__ Zpart, int i0) {
    const int lane = threadIdx.x;
    const int l15  = lane & 15;
    const int hi   = lane >> 4;
    const int m0   = blockIdx.x * 16;        // tile of previous rows
    const int s    = blockIdx.y;             // L-slice
    const int kb   = s * KSLICE;

    const float* arow = Q     + (size_t)(m0 + l15) * GS_L + kb + 2 * hi;
    const float* brow = Xorig + (size_t)(i0 + l15) * GS_L + kb + 2 * hi;

    v8f acc = {};
    for (int kk = 0; kk < KSLICE; kk += 4) {
        __builtin_prefetch(arow + kk + 256, 0, 3);   // near-cache prefetch
        v2f a = *(const v2f*)(arow + kk);
        v2f b = *(const v2f*)(brow + kk);
        acc = __builtin_amdgcn_wmma_f32_16x16x4_f32(
            false, a, false, b, (short)0, acc, false, false);
    }

    float* zp = Zpart + (size_t)s * GS_K * 16;
    const int mbase = m0 + (hi ? 8 : 0);
    #pragma unroll
    for (int r = 0; r < 8; ++r)
        zp[(size_t)(mbase + r) * 16 + l15] = acc[r];
}

// ---------------------------------------------------------------------------
// Combine partials and form TRANSPOSED coefficients:
//   Ct[j*GS_K + m] = (sum_s Zpart[s][m][j]) / z2[m],  m < i0, j < 16
// Transposed layout makes the update-GEMM A-fragment a contiguous b64 load.
// ---------------------------------------------------------------------------
__global__ void __launch_bounds__(256)
gs_panel_coef(const float* __restrict__ Zpart, const float* __restrict__ z2,
              float* __restrict__ Ct, int i0) {
    const int idx = blockIdx.x * 256 + threadIdx.x;
    if (idx >= i0 * 16) return;
    const int m = idx >> 4;
    const int j = idx & 15;
    float s = 0.f;
    #pragma unroll
    for (int t = 0; t < LSPLIT; ++t)
        s += Zpart[((size_t)t * GS_K + m) * 16 + j];
    Ct[(size_t)j * GS_K + m] = s / z2[m];
}

// ---------------------------------------------------------------------------
// Kernel B: Q_panel(16 x L) -= Ct(16 x i0) @ Q_prev(i0 x L)
// One wave per 32-column tile; A fragment reused for two WMMAs (register
// blocking over N) -> ~1 b64 + 4 b32 loads per 2 WMMAs.
// ---------------------------------------------------------------------------
__global__ void __launch_bounds__(32)
gs_panel_update(float* __restrict__ Q, const float* __restrict__ Ct, int i0) {
    const int lane = threadIdx.x;
    const int l15  = lane & 15;
    const int hi   = lane >> 4;
    const int n0   = blockIdx.x * 32;

    const float* arow = Ct + (size_t)l15 * GS_K + 2 * hi;

    v8f acc0 = {}, acc1 = {};
    for (int kk = 0; kk < i0; kk += 4) {
        v2f a = *(const v2f*)(arow + kk);            // contiguous coeff pair
        const int k0 = kk + 2 * hi;
        v2f b0, b1;
        b0.x = Q[(size_t)(k0    ) * GS_L + n0 + l15];
        b0.y = Q[(size_t)(k0 + 1) * GS_L + n0 + l15];
        b1.x = Q[(size_t)(k0    ) * GS_L + n0 + 16 + l15];
        b1.y = Q[(size_t)(k0 + 1) * GS_L + n0 + 16 + l15];
        acc0 = __builtin_amdgcn_wmma_f32_16x16x4_f32(
            false, a, false, b0, (short)0, acc0, false, false);
        acc1 = __builtin_amdgcn_wmma_f32_16x16x4_f32(
            false, a, false, b1, (short)0, acc1, false, false);
    }

    const int mrow = i0 + (hi ? 8 : 0);
    #pragma unroll
    for (int r = 0; r < 8; ++r) {
        Q[(size_t)(mrow + r) * GS_L + n0 + l15]      -= acc0[r];
        Q[(size_t)(mrow + r) * GS_L + n0 + 16 + l15] -= acc1[r];
    }
}

// ---------------------------------------------------------------------------
// Intra-panel sequential CGS (coefficients vs ORIGINAL x rows, matching the
// reference scan). Produces z2 (squared norms) of finalized rows.
// ---------------------------------------------------------------------------
__global__ void __launch_bounds__(256)
gs_panel_intra(float* __restrict__ Q, const float* __restrict__ Xorig,
               float* __restrict__ z2, int i0) {
    __shared__ float red[256];
    __shared__ float coef[PANEL];
    __shared__ float z2loc[PANEL];
    const int t = threadIdx.x;

    for (int j = 0; j < PANEL; ++j) {
        const float* xj = Xorig + (size_t)(i0 + j) * GS_L;
        for (int k = 0; k < j; ++k) {
            const float* yk = Q + (size_t)(i0 + k) * GS_L;
            float s = 0.f;
            for (int c = t; c < GS_L; c += 256) s += yk[c] * xj[c];
            red[t] = s; __syncthreads();
            for (int off = 128; off > 0; off >>= 1) {
                if (t < off) red[t] += red[t + off];
                __syncthreads();
            }
            if (t == 0) coef[k] = red[0] / z2loc[k];
            __syncthreads();
        }
        float* qj = Q + (size_t)(i0 + j) * GS_L;
        for (int c = t; c < GS_L; c += 256) {
            float v = qj[c];
            for (int k = 0; k < j; ++k)
                v -= coef[k] * Q[(size_t)(i0 + k) * GS_L + c];
            qj[c] = v;
        }
        __syncthreads();
        float s = 0.f;
        for (int c = t; c < GS_L; c += 256) { float v = qj[c]; s += v * v; }
        red[t] = s; __syncthreads();
        for (int off = 128; off > 0; off >>= 1) {
            if (t < off) red[t] += red[t + off];
            __syncthreads();
        }
        if (t == 0) { z2loc[j] = red[0]; z2[i0 + j] = red[0]; }
        __syncthreads();
    }
}

// ---------------------------------------------------------------------------
__global__ void __launch_bounds__(256)
gs_copy(const float* __restrict__ src, float* __restrict__ dst) {
    const size_t i = (size_t)blockIdx.x * 256 + threadIdx.x;
    dst[i] = src[i];
}

__global__ void __launch_bounds__(256)
gs_normalize(float* __restrict__ Q, const float* __restrict__ z2) {
    const size_t i = (size_t)blockIdx.x * 256 + threadIdx.x;
    const int row = (int)(i / GS_L);
    Q[i] *= rsqrtf(z2[row]);
}

// ---------------------------------------------------------------------------
extern "C" void kernel_launch(void* const* d_in, const int* in_sizes, int n_in,
                              void* d_out, int out_size, void* d_ws, size_t ws_size,
                              hipStream_t stream) {
    (void)in_sizes; (void)n_in; (void)out_size; (void)ws_size;
    const float* x = (const float*)d_in[0];
    float*       Q = (float*)d_out;

    float* z2    = (float*)d_ws;                 // GS_K floats
    float* Ct    = z2 + GS_K;                    // 16 * GS_K floats (transposed coeffs)
    float* Zpart = Ct + (size_t)PANEL * GS_K;    // LSPLIT * GS_K * 16 floats

    const size_t total = (size_t)GS_K * GS_L;
    gs_copy<<<(unsigned)(total / 256), 256, 0, stream>>>(x, Q);

    for (int p = 0; p < GS_K / PANEL; ++p) {
        const int i0 = p * PANEL;
        if (i0 > 0) {
            gs_panel_corr <<<dim3(i0 / 16, LSPLIT), 32, 0, stream>>>(Q, x, Zpart, i0);
            gs_panel_coef <<<(i0 * 16 + 255) / 256, 256, 0, stream>>>(Zpart, z2, Ct, i0);
            gs_panel_update<<<GS_L / 32, 32, 0, stream>>>(Q, Ct, i0);
        }
        gs_panel_intra<<<1, 256, 0, stream>>>(Q, x, z2, i0);
    }

    gs_normalize<<<(unsigned)(total / 256), 256, 0, stream>>>(Q, z2);
}